// pairwise_ranking_objective_35725537968395
// MI455X (gfx1250) — compile-verified
//
#include <hip/hip_runtime.h>
#include <math.h>

// Pairwise ranking hinge loss vs 100k contrastive rows.
// Roofline: 410 MB streamed @ 23.3 TB/s => ~17.6 us floor; 1 flop/byte useful.
// Main loop uses V_WMMA_F32_16X16X4_F32: Gram trick (A fed as both A and B)
// yields row norms on the diagonal; a second accumulator with m-hat broadcast
// across B columns yields dot(c_row, m_hat). 1x b128 load / lane / 8 columns.

#define MARGIN 0.7f
#define EPSV   1e-8f
#define DDIM   1024

typedef __attribute__((ext_vector_type(2))) float v2f;
typedef __attribute__((ext_vector_type(4))) float v4f;
typedef __attribute__((ext_vector_type(8))) float v8f;

// ---------------------------------------------------------------- kernel 1 --
// ws[0]        = MARGIN - cos(m, a)
// ws[16..16+D) = m / max(||m||, eps)
__global__ __launch_bounds__(256) void prep_kernel(const float* __restrict__ m,
                                                   const float* __restrict__ a,
                                                   float* __restrict__ ws) {
    __shared__ float smm[256], sma[256], saa[256];
    const int t = threadIdx.x;
    float mm = 0.f, ma = 0.f, aa = 0.f;
    for (int i = t; i < DDIM; i += 256) {
        const float mv = m[i], av = a[i];
        mm = fmaf(mv, mv, mm);
        ma = fmaf(mv, av, ma);
        aa = fmaf(av, av, aa);
    }
    smm[t] = mm; sma[t] = ma; saa[t] = aa;
    __syncthreads();
    for (int s = 128; s > 0; s >>= 1) {
        if (t < s) { smm[t] += smm[t + s]; sma[t] += sma[t + s]; saa[t] += saa[t + s]; }
        __syncthreads();
    }
    const float nm = fmaxf(sqrtf(smm[0]), EPSV);
    const float na = fmaxf(sqrtf(saa[0]), EPSV);
    if (t == 0) ws[0] = MARGIN - sma[0] / (nm * na);
    const float inv = 1.0f / nm;
    for (int i = t; i < DDIM; i += 256) ws[16 + i] = m[i] * inv;
}

// ---------------------------------------------------------------- kernel 2 --
// One wave per 16-row tile. accG = A*A^T (diag = ||row||^2), accM = A*Bm
// (Bm = m_hat broadcast into all 16 columns => every column holds dot(row,m_hat)).
__global__ __launch_bounds__(256) void tile_kernel(const float* __restrict__ c,
                                                   const float* __restrict__ ws,
                                                   float* __restrict__ partial,
                                                   int K, int NT) {
    const int lane = threadIdx.x & 31;
    const int wave = threadIdx.x >> 5;
    const int tile = blockIdx.x * 8 + wave;
    if (tile >= NT) return;               // wave-uniform: EXEC stays all-ones

    const float  mms  = ws[0];            // MARGIN - s_match (scalar load)
    const float* mhat = ws + 16;

    const int h = lane >> 4;              // half-wave selects K pair {2h, 2h+1}
    const int r = lane & 15;              // row within tile
    const int row  = tile * 16 + r;
    const int rowc = (row < K) ? row : (K - 1);   // clamp tail loads (safe)

    const float* crow = c + (size_t)rowc * DDIM + 4 * h;
    const float* mrow = mhat + 4 * h;

    v8f accG = {};                        // Gram accumulator (norms on diag)
    v8f accM = {};                        // dot-with-m accumulator

    // Per 8 columns: one b128 per lane covers two K-chunks for this half-wave.
    // K-label permutation is identical for A and B operands, so sums are exact.
    #pragma unroll 2
    for (int k = 0; k < DDIM; k += 8) {
        const v4f av = *(const v4f*)(crow + k);
        const v4f bv = *(const v4f*)(mrow + k);
        v2f a0 = {av.x, av.y};
        v2f a1 = {av.z, av.w};
        v2f b0 = {bv.x, bv.y};
        v2f b1 = {bv.z, bv.w};
        accG = __builtin_amdgcn_wmma_f32_16x16x4_f32(false, a0, false, a0, (short)0, accG, false, false);
        accM = __builtin_amdgcn_wmma_f32_16x16x4_f32(false, a0, false, b0, (short)0, accM, false, false);
        accG = __builtin_amdgcn_wmma_f32_16x16x4_f32(false, a1, false, a1, (short)0, accG, false, false);
        accM = __builtin_amdgcn_wmma_f32_16x16x4_f32(false, a1, false, b1, (short)0, accM, false, false);
    }

    // C/D layout: VGPR v, lanes 0-15 -> (M=v, N=lane); lanes 16-31 -> (M=v+8, N=lane-16).
    // Diagonal ||c_M||^2 for M = v + 8h lives at lane v + 24h, VGPR v.
    // accM[v] is column-uniform = dot(c_{v+8h}, m_hat).
    float psum = 0.f;
    #pragma unroll
    for (int v = 0; v < 8; ++v) {
        const int src  = v + 24 * h;
        const float nsq  = __shfl(accG[v], src, 32);
        const float dm   = accM[v];
        const float cosv = dm / fmaxf(sqrtf(nsq), EPSV);
        const float hin  = fmaxf(0.f, mms + cosv);
        const int   rr   = tile * 16 + v + 8 * h;
        if (r == 0 && rr < K) psum += hin;   // lane 0: rows 0..7, lane 16: rows 8..15
    }
    #pragma unroll
    for (int off = 16; off; off >>= 1) psum += __shfl_xor(psum, off, 32);
    if (lane == 0) partial[tile] = psum;
}

// ---------------------------------------------------------------- kernel 3 --
// Deterministic fixed-order reduction of per-tile partials.
__global__ __launch_bounds__(256) void reduce_kernel(const float* __restrict__ partial,
                                                     int n, float* __restrict__ out) {
    __shared__ float buf[256];
    const int t = threadIdx.x;
    float s = 0.f;
    for (int i = t; i < n; i += 256) s += partial[i];
    buf[t] = s;
    __syncthreads();
    for (int k = 128; k > 0; k >>= 1) {
        if (t < k) buf[t] += buf[t + k];
        __syncthreads();
    }
    if (t == 0) out[0] = buf[0];
}

// ------------------------------------------------------------------- launch --
extern "C" void kernel_launch(void* const* d_in, const int* in_sizes, int n_in,
                              void* d_out, int out_size, void* d_ws, size_t ws_size,
                              hipStream_t stream) {
    const float* midi  = (const float*)d_in[0];   // (1, 1024)
    const float* audio = (const float*)d_in[1];   // (1, 1024)
    const float* contr = (const float*)d_in[2];   // (K, 1024)
    float* out = (float*)d_out;
    float* ws  = (float*)d_ws;

    const int K  = in_sizes[2] / DDIM;            // 100000
    const int NT = (K + 15) / 16;                 // 6250 tiles

    float* partial = ws + 2048;                   // ws: [0]=mms, [16..]=m_hat, [2048..]=partials

    prep_kernel<<<1, 256, 0, stream>>>(midi, audio, ws);
    const int blocks = (NT + 7) / 8;              // 8 waves (tiles) per block
    tile_kernel<<<blocks, 256, 0, stream>>>(contr, ws, partial, K, NT);
    reduce_kernel<<<1, 256, 0, stream>>>(partial, NT, out);
}